// MessagePassingAttentionPool_43052752175240
// MI455X (gfx1250) — compile-verified
//
#include <hip/hip_runtime.h>
#include <hip/hip_bf16.h>
#include <math.h>

// Problem constants (match reference)
#define N_NODES 16384   // B*K
#define BMOL    1024
#define KCONF   16
#define CDIM    128
#define HOUT    256     // HEADS*C
#define NHEAD   2

typedef float v2f __attribute__((ext_vector_type(2)));
typedef float v8f __attribute__((ext_vector_type(8)));

#if defined(__has_builtin)
#if __has_builtin(__builtin_amdgcn_wmma_f32_16x16x4_f32)
#define HAS_WMMA_F32 1
#endif
#endif

// ---------------------------------------------------------------------------
// Y[mbase+0..31, 0..255] = act(X[32,K] @ W[K,256] + bias)
// One block per 32-row tile (2 WMMA M-tiles -> each B fragment feeds 2 WMMAs),
// 8 waves, each wave owns two 16-col tiles. fp32 WMMA 16x16x4.
// ACT: 0 = identity, 1 = LeakyReLU(0.01)
// ---------------------------------------------------------------------------
template<int K, int ACT>
__global__ __launch_bounds__(256)
void gemm16_wmma(const float* __restrict__ X, const float* __restrict__ W,
                 const float* __restrict__ bias, float* __restrict__ Y)
{
    constexpr int NOUT = 256;
    constexpr int LDK  = K + 4;          // pad (16B-aligned rows, no 16-way conflicts)
    __shared__ float sx[32 * LDK];

    const int mbase = blockIdx.x << 5;   // 32 rows per block
    const int tid   = threadIdx.x;

    // Stage X tile as float4: global_load_b128 -> ds_store_b128
    constexpr int NV4 = 32 * K / 4;
    for (int i = tid; i < NV4; i += 256) {
        const int r  = i / (K / 4);
        const int c4 = i - r * (K / 4);
        const float4 v = ((const float4*)(X + (size_t)(mbase + r) * K))[c4];
        *(float4*)(&sx[r * LDK + c4 * 4]) = v;
    }
    __syncthreads();

    const int lane  = tid & 31;
    const int wave  = tid >> 5;
    const int arow  = lane & 15;            // M row (A) / N col (B,D) within tile
    const int khalf = (lane >> 4) << 1;     // lanes 0-15 -> K{0,1}, 16-31 -> K{2,3}
    const int c0    = wave << 5;            // two col tiles: [c0, c0+16)

    v8f acc[2][2] = {{{}, {}}, {{}, {}}};

    for (int k = 0; k < K; k += 4) {
        const float* wp = W + (size_t)(k + khalf) * NOUT;
#ifdef HAS_WMMA_F32
        v2f a0, a1, b0, b1;
        a0.x = sx[arow * LDK + k + khalf];
        a0.y = sx[arow * LDK + k + khalf + 1];
        a1.x = sx[(16 + arow) * LDK + k + khalf];
        a1.y = sx[(16 + arow) * LDK + k + khalf + 1];
        b0.x = wp[c0 + arow];
        b0.y = wp[NOUT + c0 + arow];
        b1.x = wp[16 + c0 + arow];
        b1.y = wp[NOUT + 16 + c0 + arow];
        if (k + 4 < K)  // pull next W k-slab toward cache (global_prefetch_b8)
            __builtin_prefetch((const void*)(wp + 4 * NOUT + c0), 0, 1);
        acc[0][0] = __builtin_amdgcn_wmma_f32_16x16x4_f32(
                        false, a0, false, b0, (short)0, acc[0][0], false, false);
        acc[0][1] = __builtin_amdgcn_wmma_f32_16x16x4_f32(
                        false, a0, false, b1, (short)0, acc[0][1], false, false);
        acc[1][0] = __builtin_amdgcn_wmma_f32_16x16x4_f32(
                        false, a1, false, b0, (short)0, acc[1][0], false, false);
        acc[1][1] = __builtin_amdgcn_wmma_f32_16x16x4_f32(
                        false, a1, false, b1, (short)0, acc[1][1], false, false);
#else
        // VALU fallback (same D-fragment layout); only if builtin missing.
        for (int m = 0; m < 2; ++m)
            for (int i = 0; i < 8; ++i) {
                const int r   = m * 16 + i + ((lane >> 4) << 3);
                const int col = c0 + (lane & 15);
                float s0 = 0.f, s1 = 0.f;
                for (int j = 0; j < 4; ++j) {
                    const float xv = sx[r * LDK + k + j];
                    s0 += xv * W[(size_t)(k + j) * NOUT + col];
                    s1 += xv * W[(size_t)(k + j) * NOUT + col + 16];
                }
                acc[m][0][i] += s0;
                acc[m][1][i] += s1;
            }
        (void)wp;
#endif
    }

    // D fragment: VGPR i -> rows {i, i+8} split across lane halves, col = lane&15
    const int col = c0 + (lane & 15);
    const float bc0 = bias[col];
    const float bc1 = bias[col + 16];
    for (int m = 0; m < 2; ++m)
        for (int i = 0; i < 8; ++i) {
            const int r = m * 16 + i + ((lane >> 4) << 3);
            float v0 = acc[m][0][i] + bc0;
            float v1 = acc[m][1][i] + bc1;
            if (ACT) {
                v0 = (v0 >= 0.f) ? v0 : 0.01f * v0;
                v1 = (v1 >= 0.f) ? v1 : 0.01f * v1;
            }
            Y[(size_t)(mbase + r) * NOUT + col]      = v0;
            Y[(size_t)(mbase + r) * NOUT + col + 16] = v1;
        }
}

// ---------------------------------------------------------------------------
// Dense per-segment GATv2 attention. One block per 16-node segment.
// xl/xr: [N, HEADS*C]; att: [HEADS*C]; out: [N, C] (mean over heads + bias)
// ---------------------------------------------------------------------------
__global__ __launch_bounds__(256)
void gat_attention(const float* __restrict__ xl, const float* __restrict__ xr,
                   const float* __restrict__ att, const float* __restrict__ bias,
                   float* __restrict__ out)
{
    __shared__ float sxl[16 * HOUT];
    __shared__ float sxr[16 * HOUT];
    __shared__ float satt[HOUT];
    __shared__ float sbias[CDIM];
    __shared__ float slog[NHEAD * 16 * 16];   // [h][src][dst], reused as alpha

    const int seg = blockIdx.x;
    const int tid = threadIdx.x;
    const size_t base4 = (size_t)seg * 16 * HOUT / 4;

    // float4-vectorized staging (16*256/4 = 1024 float4 per array)
    for (int i = tid; i < 16 * HOUT / 4; i += 256) {
        ((float4*)sxl)[i] = ((const float4*)xl)[base4 + i];
        ((float4*)sxr)[i] = ((const float4*)xr)[base4 + i];
    }
    if (tid < HOUT) satt[tid] = att[tid];
    if (tid < CDIM) sbias[tid] = bias[tid];
    __syncthreads();

    // logits[h][s][d] = sum_c leaky(xl[s,h,c] + xr[d,h,c], 0.2) * att[h,c]
    for (int t = tid; t < NHEAD * 256; t += 256) {
        const int h = t >> 8;
        const int s = (t >> 4) & 15;
        const int d = t & 15;
        const float* pl = &sxl[s * HOUT + h * CDIM];
        const float* pr = &sxr[d * HOUT + h * CDIM];
        const float* pa = &satt[h * CDIM];
        float acc = 0.f;
        for (int c = 0; c < CDIM; ++c) {
            float v = pl[c] + pr[c];
            v = (v >= 0.f) ? v : 0.2f * v;
            acc += v * pa[c];
        }
        slog[t] = acc;
    }
    __syncthreads();

    // softmax over the 16 sources for each (h, d): 32 independent softmaxes
    if (tid < NHEAD * 16) {
        const int h = tid >> 4;
        const int d = tid & 15;
        float m = -3.4e38f;
        for (int s = 0; s < 16; ++s) {
            const float v = slog[h * 256 + s * 16 + d];
            m = (v > m) ? v : m;
        }
        float den = 0.f;
        for (int s = 0; s < 16; ++s) {
            const float e = expf(slog[h * 256 + s * 16 + d] - m);
            slog[h * 256 + s * 16 + d] = e;   // stash exp
            den += e;
        }
        const float inv = 1.f / den;
        for (int s = 0; s < 16; ++s)
            slog[h * 256 + s * 16 + d] *= inv;
    }
    __syncthreads();

    // out[d,c] = mean_h( sum_s alpha[h,s,d] * xl[s,h,c] ) + bias[c]
    for (int t = tid; t < 16 * CDIM; t += 256) {
        const int d = t >> 7;
        const int c = t & 127;
        float acc = 0.f;
        for (int h = 0; h < NHEAD; ++h)
            for (int s = 0; s < 16; ++s)
                acc += slog[h * 256 + s * 16 + d] * sxl[s * HOUT + h * CDIM + c];
        out[(size_t)(seg * 16 + d) * CDIM + c] = 0.5f * acc + sbias[c];
    }
}

// ---------------------------------------------------------------------------
// h_gate[n] = g2[n,:] . gW3 + gb3 ; one wave32 per node
// ---------------------------------------------------------------------------
__global__ __launch_bounds__(256)
void gate_dot(const float* __restrict__ g2, const float* __restrict__ gW3,
              const float* __restrict__ gb3, float* __restrict__ hg)
{
    const int node = blockIdx.x * 8 + (threadIdx.x >> 5);
    const int lane = threadIdx.x & 31;
    float acc = 0.f;
    for (int c = lane; c < HOUT; c += 32)
        acc += g2[(size_t)node * HOUT + c] * gW3[c];
    for (int m = 16; m >= 1; m >>= 1)
        acc += __shfl_xor(acc, m, 32);
    if (lane == 0) hg[node] = acc + gb3[0];
}

// ---------------------------------------------------------------------------
// Per-molecule softmax over 16 gate logits + weighted pooling of x.
// out_pool: [B, C], out_scores: [N]
// ---------------------------------------------------------------------------
__global__ __launch_bounds__(128)
void pool_kernel(const float* __restrict__ x, const float* __restrict__ hg,
                 float* __restrict__ out_pool, float* __restrict__ out_scores)
{
    __shared__ float sg[16];
    __shared__ float sa[16];
    __shared__ float smx, sden;

    const int mol = blockIdx.x;
    const int tid = threadIdx.x;

    if (tid < 16) sg[tid] = hg[mol * 16 + tid];
    __syncthreads();
    if (tid == 0) {
        float m = -3.4e38f;
        for (int k = 0; k < 16; ++k) m = (sg[k] > m) ? sg[k] : m;
        smx = m;
    }
    __syncthreads();
    if (tid < 16) sa[tid] = expf(sg[tid] - smx);
    __syncthreads();
    if (tid == 0) {
        float d = 0.f;
        for (int k = 0; k < 16; ++k) d += sa[k];
        sden = d;
    }
    __syncthreads();
    if (tid < 16) {
        const float sc = sa[tid] / sden;
        sa[tid] = sc;
        out_scores[mol * 16 + tid] = sc;
    }
    __syncthreads();

    // tid == channel c
    float acc = 0.f;
    for (int k = 0; k < 16; ++k)
        acc += sa[k] * x[(size_t)(mol * 16 + k) * CDIM + tid];
    out_pool[(size_t)mol * CDIM + tid] = acc;
}

// ---------------------------------------------------------------------------
extern "C" void kernel_launch(void* const* d_in, const int* in_sizes, int n_in,
                              void* d_out, int out_size, void* d_ws, size_t ws_size,
                              hipStream_t stream)
{
    (void)in_sizes; (void)n_in; (void)out_size; (void)ws_size;

    const float* H        = (const float*)d_in[0];
    // d_in[1..3] = batch / edge_src / edge_dst : graph structure is implicit
    const float* Wl       = (const float*)d_in[4];
    const float* bl       = (const float*)d_in[5];
    const float* Wr       = (const float*)d_in[6];
    const float* br       = (const float*)d_in[7];
    const float* att      = (const float*)d_in[8];
    const float* gat_bias = (const float*)d_in[9];
    const float* gW1      = (const float*)d_in[10];
    const float* gb1      = (const float*)d_in[11];
    const float* gW2      = (const float*)d_in[12];
    const float* gb2      = (const float*)d_in[13];
    const float* gW3      = (const float*)d_in[14];
    const float* gb3      = (const float*)d_in[15];

    // workspace layout (floats)
    float* ws = (float*)d_ws;
    float* xl = ws;                                  // [N, 256]  (aliased as g1)
    float* xr = xl + (size_t)N_NODES * HOUT;         // [N, 256]  (aliased as g2)
    float* x1 = xr + (size_t)N_NODES * HOUT;         // [N, 128]
    float* x2 = x1 + (size_t)N_NODES * CDIM;         // [N, 128]
    float* hg = x2 + (size_t)N_NODES * CDIM;         // [N]

    const int GEMM_BLKS = N_NODES / 32;              // 32 rows per block
    const float* xcur = H;
    float* xnext[2] = { x1, x2 };

    for (int l = 0; l < 2; ++l) {
        gemm16_wmma<CDIM, 0><<<GEMM_BLKS, 256, 0, stream>>>(
            xcur, Wl + (size_t)l * CDIM * HOUT, bl + (size_t)l * HOUT, xl);
        gemm16_wmma<CDIM, 0><<<GEMM_BLKS, 256, 0, stream>>>(
            xcur, Wr + (size_t)l * CDIM * HOUT, br + (size_t)l * HOUT, xr);
        gat_attention<<<BMOL, 256, 0, stream>>>(
            xl, xr, att + (size_t)l * HOUT, gat_bias + (size_t)l * CDIM, xnext[l]);
        xcur = xnext[l];
    }

    // gate MLP: g1 = leaky(x2@gW1+gb1), g2 = leaky(g1@gW2+gb2)
    gemm16_wmma<CDIM, 1><<<GEMM_BLKS, 256, 0, stream>>>(x2, gW1, gb1, xl);
    gemm16_wmma<HOUT, 1><<<GEMM_BLKS, 256, 0, stream>>>(xl, gW2, gb2, xr);
    gate_dot<<<N_NODES / 8, 256, 0, stream>>>(xr, gW3, gb3, hg);

    float* out_pool   = (float*)d_out;                    // [B, C]
    float* out_scores = out_pool + (size_t)BMOL * CDIM;   // [N]
    pool_kernel<<<BMOL, 128, 0, stream>>>(x2, hg, out_pool, out_scores);
}